// GraphQNetwork_70025146794593
// MI455X (gfx1250) — compile-verified
//
#include <hip/hip_runtime.h>

typedef __attribute__((ext_vector_type(2))) float v2f;
typedef __attribute__((ext_vector_type(8))) float v8f;

constexpr int GPB   = 16;           // graphs per block (16*9 = 144 = 9 WMMA M-tiles)
constexpr int NPG   = 9;
constexpr int NODES = GPB * NPG;    // 144
constexpr int EPG   = 36;
constexpr int EDG   = GPB * EPG;    // 576
constexpr int FIN   = 16;
constexpr int H1    = 32;
constexpr int H2    = 64;
constexpr int TPB   = 256;          // 8 waves (wave32)

// ---- LDS layout (single buffer => compile-time byte offsets for async copies)
constexpr unsigned OFF_X    = 0;                        //  9216 B : input tile
constexpr unsigned OFF_HPRE = OFF_X    + NODES*FIN*4;   // 36864 B : x@W (both layers)
constexpr unsigned OFF_A1   = OFF_HPRE + NODES*H2*4;    // 18432 B : layer-1 activations
constexpr unsigned OFF_A2   = OFF_A1   + NODES*H1*4;    // 36864 B : layer-2 activations
constexpr unsigned OFF_CTX  = OFF_A2   + NODES*H2*4;    //  4096 B : graph mean pool
constexpr unsigned OFF_HID  = OFF_CTX  + GPB*H2*4;      //  8192 B : fc1 hidden
constexpr unsigned OFF_DINV = OFF_HID  + 2*GPB*H2*4;    //   576 B
constexpr unsigned OFF_DEG  = OFF_DINV + NODES*4;       //   576 B
constexpr unsigned OFF_ESD  = OFF_DEG  + NODES*4;       //  2304 B : esrc stage -> src|dst<<16
constexpr unsigned OFF_ENM  = OFF_ESD  + EDG*4;         //  2304 B : edst stage -> norm
constexpr unsigned SMEM_SZ  = OFF_ENM  + EDG*4;         // 119424 B total (< 320 KB/WGP)

__device__ __forceinline__ v8f wmma_f32_k4(v2f a, v2f b, v8f c) {
  // V_WMMA_F32_16X16X4_F32 : D = A(16x4,f32) * B(4x16,f32) + C(16x16,f32)
  return __builtin_amdgcn_wmma_f32_16x16x4_f32(false, a, false, b, (short)0, c,
                                               false, false);
}

// 16-byte async global -> LDS copy (ASYNCcnt-tracked, bypasses VGPR data path)
__device__ __forceinline__ void async_cp16(unsigned lds_off, const void* g) {
  asm volatile("global_load_async_to_lds_b128 %0, %1, off"
               :: "v"(lds_off), "v"((unsigned long long)(size_t)g) : "memory");
}
__device__ __forceinline__ void wait_async0() {
  asm volatile("s_wait_asynccnt 0x0" ::: "memory");
}

__global__ __launch_bounds__(TPB) void gqn_fused(
    const float* __restrict__ x,
    const int*   __restrict__ esrc,
    const int*   __restrict__ edst,
    const float* __restrict__ W1, const float* __restrict__ b1,
    const float* __restrict__ W2, const float* __restrict__ b2,
    const float* __restrict__ fc1w, const float* __restrict__ fc1b,
    const float* __restrict__ fc2w, const float* __restrict__ fc2b,
    float* __restrict__ out)
{
  __shared__ __align__(16) unsigned char smem[SMEM_SZ];
  float* sX    = (float*)(smem + OFF_X);
  float* sHpre = (float*)(smem + OFF_HPRE);
  float* sA1   = (float*)(smem + OFF_A1);
  float* sA2   = (float*)(smem + OFF_A2);
  float* sCtx  = (float*)(smem + OFF_CTX);
  float* sHid  = (float*)(smem + OFF_HID);
  float* sDinv = (float*)(smem + OFF_DINV);
  int*   sDeg  = (int*)  (smem + OFF_DEG);
  int*   sESD  = (int*)  (smem + OFF_ESD);   // staged esrc, repacked to src|dst<<16
  float* sENm  = (float*)(smem + OFF_ENM);   // staged edst, repacked to norm
  int*   sED   = (int*)  (smem + OFF_ENM);   // int view of staged edst

  const int tid  = (int)threadIdx.x;
  const int lane = tid & 31;
  const int wave = tid >> 5;
  const int g0   = (int)blockIdx.x * GPB;  // first graph of this block
  const int n0   = g0 * NPG;               // first node
  const int e0   = g0 * EPG;               // first edge

  // ---- async-stage x tile + both edge arrays into LDS; clear degrees -------
  {
    const char* xg = (const char*)(x + (size_t)n0 * FIN);
    for (int i = tid; i < NODES * FIN / 4; i += TPB)     // 576 x 16B
      async_cp16(OFF_X + i * 16u, xg + i * 16);
    const char* sg = (const char*)(esrc + e0);
    const char* dg = (const char*)(edst + e0);
    for (int i = tid; i < EDG / 4; i += TPB) {           // 144 x 16B each
      async_cp16(OFF_ESD + i * 16u, sg + i * 16);
      async_cp16(OFF_ENM + i * 16u, dg + i * 16);
    }
    for (int i = tid; i < NODES; i += TPB) sDeg[i] = 0;
    wait_async0();
  }
  __syncthreads();

  // ---- degrees (in-degree on dst; +1 self loop) -----------------------------
  for (int e = tid; e < EDG; e += TPB)
    atomicAdd(&sDeg[sED[e] - n0], 1);
  __syncthreads();
  for (int i = tid; i < NODES; i += TPB)
    sDinv[i] = rsqrtf((float)(sDeg[i] + 1));
  __syncthreads();
  // in-place repack: sESD <- src|dst<<16, sENm <- dinv[src]*dinv[dst]
  for (int e = tid; e < EDG; e += TPB) {
    const int s = sESD[e] - n0;            // 0..143
    const int d = sED[e]  - n0;            // 0..143
    sESD[e] = s | (d << 16);
    sENm[e] = sDinv[s] * sDinv[d];
  }
  __syncthreads();

  // ---- GEMM1: sHpre[144x32] = sX[144x16] @ W1[16x32]  (WMMA fp32) ----------
  {
    const int khalf = (lane >> 4) << 1;          // K offset 0 / 2 per lane half
    const int mr    = lane & 15;
    for (int t = wave; t < 18; t += 8) {         // 9 Mtiles x 2 Ntiles
      const int arow = (t >> 1) * 16 + mr;
      const int bcol = (t & 1) * 16 + mr;
      v8f c = {};
      #pragma unroll
      for (int k0 = 0; k0 < FIN; k0 += 4) {
        const int kb = k0 + khalf;
        v2f a = *(const v2f*)(sX + arow * FIN + kb);
        v2f b; b.x = W1[kb * H1 + bcol]; b.y = W1[(kb + 1) * H1 + bcol];
        c = wmma_f32_k4(a, b, c);
      }
      const int rb = (t >> 1) * 16 + ((lane >> 4) << 3);
      #pragma unroll
      for (int v = 0; v < 8; ++v) sHpre[(rb + v) * H1 + bcol] = c[v];
    }
  }
  __syncthreads();

  // ---- Aggregation 1: self-loop init, edge scatter (ds_add_f32), bias+ReLU -
  for (int i = tid; i < NODES * H1; i += TPB) {
    const float dv = sDinv[i >> 5];
    sA1[i] = dv * dv * sHpre[i];
  }
  __syncthreads();
  for (int i = tid; i < EDG * H1; i += TPB) {
    const int e = i >> 5, f = i & 31;
    const int sd = sESD[e];
    atomicAdd(&sA1[(sd >> 16) * H1 + f],
              sENm[e] * sHpre[(sd & 0xffff) * H1 + f]);
  }
  __syncthreads();
  for (int i = tid; i < NODES * H1; i += TPB)
    sA1[i] = fmaxf(sA1[i] + b1[i & 31], 0.0f);
  __syncthreads();

  // ---- GEMM2: sHpre[144x64] = sA1[144x32] @ W2[32x64] ----------------------
  {
    const int khalf = (lane >> 4) << 1;
    const int mr    = lane & 15;
    for (int t = wave; t < 36; t += 8) {         // 9 Mtiles x 4 Ntiles
      const int arow = (t >> 2) * 16 + mr;
      const int bcol = (t & 3) * 16 + mr;
      v8f c = {};
      #pragma unroll
      for (int k0 = 0; k0 < H1; k0 += 4) {
        const int kb = k0 + khalf;
        v2f a = *(const v2f*)(sA1 + arow * H1 + kb);
        v2f b; b.x = W2[kb * H2 + bcol]; b.y = W2[(kb + 1) * H2 + bcol];
        c = wmma_f32_k4(a, b, c);
      }
      const int rb = (t >> 2) * 16 + ((lane >> 4) << 3);
      #pragma unroll
      for (int v = 0; v < 8; ++v) sHpre[(rb + v) * H2 + bcol] = c[v];
    }
  }
  __syncthreads();

  // ---- Aggregation 2 -------------------------------------------------------
  for (int i = tid; i < NODES * H2; i += TPB) {
    const float dv = sDinv[i >> 6];
    sA2[i] = dv * dv * sHpre[i];
  }
  __syncthreads();
  for (int i = tid; i < EDG * H2; i += TPB) {
    const int e = i >> 6, f = i & 63;
    const int sd = sESD[e];
    atomicAdd(&sA2[(sd >> 16) * H2 + f],
              sENm[e] * sHpre[(sd & 0xffff) * H2 + f]);
  }
  __syncthreads();
  for (int i = tid; i < NODES * H2; i += TPB)
    sA2[i] = fmaxf(sA2[i] + b2[i & 63], 0.0f);
  __syncthreads();

  // ---- Mean pool over 9 nodes per graph ------------------------------------
  for (int i = tid; i < GPB * H2; i += TPB) {
    const int g = i >> 6, f = i & 63;
    float s = 0.f;
    #pragma unroll
    for (int n = 0; n < NPG; ++n) s += sA2[(g * NPG + n) * H2 + f];
    sCtx[i] = s * (1.0f / 9.0f);
  }
  __syncthreads();

  // ---- FC1 heads via WMMA: M = 16 graphs, K = 128 (station emb ++ ctx) -----
  {
    const int khalf = (lane >> 4) << 1;
    const int st   = wave >> 2;                  // station 0 -> node 0, 1 -> node 8
    const int nt   = wave & 3;                   // N-tile of the 64 hidden units
    const int gr   = lane & 15;                  // graph row for A fragment
    const int bcol = nt * 16 + (lane & 15);
    const float* stat = sA2 + (gr * NPG + (st ? 8 : 0)) * H2;
    const float* ctx  = sCtx + gr * H2;
    v8f c = {};
    #pragma unroll
    for (int k0 = 0; k0 < 2 * H2; k0 += 4) {
      const int kb = k0 + khalf;
      v2f a;
      if (kb < H2) { a.x = stat[kb];      a.y = stat[kb + 1]; }
      else         { a.x = ctx[kb - H2];  a.y = ctx[kb - H2 + 1]; }
      v2f b; b.x = fc1w[kb * H2 + bcol]; b.y = fc1w[(kb + 1) * H2 + bcol];
      c = wmma_f32_k4(a, b, c);
    }
    const int rb = (lane >> 4) << 3;
    const float bb = fc1b[bcol];
    #pragma unroll
    for (int v = 0; v < 8; ++v)
      sHid[(st * GPB + rb + v) * H2 + bcol] = fmaxf(c[v] + bb, 0.0f);
  }
  __syncthreads();

  // ---- FC2: 64-dot per (graph, station) ------------------------------------
  if (tid < 2 * GPB) {
    const int g = tid >> 1, st = tid & 1;
    const float* h = sHid + (st * GPB + g) * H2;
    float s = 0.f;
    #pragma unroll
    for (int k = 0; k < H2; ++k) s += h[k] * fc2w[k];
    out[(size_t)(g0 + g) * 2 + st] = s + fc2b[0];
  }
}

extern "C" void kernel_launch(void* const* d_in, const int* in_sizes, int n_in,
                              void* d_out, int out_size, void* d_ws, size_t ws_size,
                              hipStream_t stream) {
  (void)n_in; (void)out_size; (void)d_ws; (void)ws_size;
  const float* x   = (const float*)d_in[0];
  const int*   ei  = (const int*)d_in[1];     // [2, E] : src row, dst row
  // d_in[2] = batch (unused; graph id == node/9)
  const float* W1  = (const float*)d_in[3];
  const float* b1  = (const float*)d_in[4];
  const float* W2  = (const float*)d_in[5];
  const float* b2  = (const float*)d_in[6];
  const float* f1w = (const float*)d_in[7];
  const float* f1b = (const float*)d_in[8];
  const float* f2w = (const float*)d_in[9];
  const float* f2b = (const float*)d_in[10];
  float* out = (float*)d_out;

  const int N = in_sizes[0] / FIN;     // 900000 nodes
  const int B = N / NPG;               // 100000 graphs
  const int E = in_sizes[1] / 2;       // 3600000 edges
  const int* src = ei;
  const int* dst = ei + E;

  const int blocks = B / GPB;          // 6250 (B divides exactly)
  gqn_fused<<<blocks, TPB, 0, stream>>>(x, src, dst, W1, b1, W2, b2,
                                        f1w, f1b, f2w, f2b, out);
}